// DeformableConv2d_22127671509599
// MI455X (gfx1250) — compile-verified
//
#include <hip/hip_runtime.h>
#include <math.h>

// ---------------------------------------------------------------------------
// Deformable Conv v2 for gfx1250 (MI455X), fp32 end-to-end via
// V_WMMA_F32_16X16X4_F32. Problem: B=4, Cin=Cout=64, H=W=160, K=3, pad=1.
//
// Pipeline (all on `stream`):
//   k_nchw_to_nhwc : x (NCHW) -> xh (NHWC) in d_ws   [26.2 MB]
//   k_offset_conv  : 27-ch conv via WMMA, writes packed per-pixel
//                    offmask[p][tap][{dy,dx,sigmoid(m)}] in d_ws [11.1 MB]
//   k_deform_conv  : bilinear sample -> LDS im2col, WMMA GEMM 64x576xN
//
// Workspace required: 37,273,600 bytes.
// ---------------------------------------------------------------------------

typedef float v2f __attribute__((ext_vector_type(2)));
typedef float v8f __attribute__((ext_vector_type(8)));

#define BQ     4
#define CIN    64
#define HH     160
#define WWID   160
#define HW     (HH * WWID)        // 25600
#define COUT   64
#define KDIM   576                // CIN * 9
#define NPIX   (BQ * HW)          // 102400
#define TILE_N 64                 // pixels per workgroup
#define CK     32                 // K-values per weight chunk (16 kp rows, 8 wmma steps)
#define NCHUNK (KDIM / CK)        // 18

// LDS float counts
#define VL_FLOATS (288 * 128)     // 288 kp rows x 64 px x float2
#define WC1_FLOATS (16 * 64)      // 16 kp rows x 32 m x float2 (offset conv)
#define WC2_FLOATS (16 * 128)     // 16 kp rows x 64 m x float2 (main conv)

// ---------------------------------------------------------------------------
// Kernel 0: NCHW -> NHWC transpose of x, LDS-tiled so both sides coalesce.
// ---------------------------------------------------------------------------
__global__ void k_nchw_to_nhwc(const float* __restrict__ x,
                               float* __restrict__ xh) {
  __shared__ float tile[64][65];
  const int p0  = blockIdx.x * 64;          // 64 flat pixels, same batch b
  const int b   = p0 / HW;
  const int hw0 = p0 - b * HW;
  const int tid = threadIdx.x;
  const int a = tid >> 6;                   // 0..3
  const int e = tid & 63;                   // 0..63
#pragma unroll
  for (int r = 0; r < 16; ++r) {
    const int c = a + r * 4;
    tile[c][e] = x[(size_t)(b * 64 + c) * HW + hw0 + e];   // coalesced read
  }
  __syncthreads();
#pragma unroll
  for (int r = 0; r < 16; ++r) {
    const int i = a + r * 4;
    xh[(size_t)(p0 + i) * 64 + e] = tile[e][i];            // coalesced write
  }
}

// ---------------------------------------------------------------------------
// Kernel 1: offset conv (27 out ch) via WMMA f32 16x16x4, then repack:
//   conv ch o<18 : tap=o>>1, comp=o&1 (dy/dx); o>=18: tap=o-18, comp=2 (sigmoid)
//   offm[p*27 + tap*3 + comp]
// 256 threads = 8 waves = 2 M-tiles (0..15 / 16..31, rows 27..31 zero-padded)
//                        x 4 N-subtiles of 16 pixels.
// ---------------------------------------------------------------------------
__global__ void k_offset_conv(const float* __restrict__ xh,
                              const float* __restrict__ ow,   // (27,576) flat [o][c*9+t]
                              const float* __restrict__ ob,   // (27,)
                              float* __restrict__ offm) {
  extern __shared__ float smem[];
  float* vl = smem;                 // VL_FLOATS  : val pairs [kp][px] float2
  float* wc = smem + VL_FLOATS;     // WC1_FLOATS : weight chunk [kp][m] float2
  const int p0  = blockIdx.x * TILE_N;
  const int tid = threadIdx.x;

  // ---- im2col sampling into LDS (regular 3x3 neighborhood, zero-pad) ----
  for (int item = tid; item < 9 * TILE_N; item += blockDim.x) {
    const int px = item & 63;
    const int t  = item >> 6;                    // 0..8, tap = ti*3+tj
    const int ti = t / 3, tj = t - 3 * (t / 3);
    const int p  = p0 + px;
    const int b  = p / HW;
    const int hw = p - b * HW;
    const int h  = hw / WWID, w = hw - (hw / WWID) * WWID;
    int y  = h + ti - 1;
    int xx = w + tj - 1;
    const float s = (y >= 0 && y < HH && xx >= 0 && xx < WWID) ? 1.f : 0.f;
    y  = min(max(y, 0), HH - 1);
    xx = min(max(xx, 0), WWID - 1);
    const float4* src =
        reinterpret_cast<const float4*>(xh + (size_t)((b * HH + y) * WWID + xx) * 64);
#pragma unroll 4
    for (int c4 = 0; c4 < 16; ++c4) {
      const float4 v = src[c4];
      int k = (c4 * 4) * 9 + t;   // K index = c*9 + t
      vl[((k >> 1) << 7) + (px << 1) + (k & 1)] = v.x * s; k += 9;
      vl[((k >> 1) << 7) + (px << 1) + (k & 1)] = v.y * s; k += 9;
      vl[((k >> 1) << 7) + (px << 1) + (k & 1)] = v.z * s; k += 9;
      vl[((k >> 1) << 7) + (px << 1) + (k & 1)] = v.w * s;
    }
  }
  __syncthreads();

  const int lane = tid & 31, wave = tid >> 5;
  const int lo = lane & 15, hi = lane >> 4;
  const int mtile = wave & 1;        // 0..1  -> out ch 0..15 / 16..31
  const int nsub  = wave >> 1;       // 0..3  -> 16-pixel subtile
  v8f acc = {};

  for (int kc = 0; kc < NCHUNK; ++kc) {
    // stage weight chunk: 16 kp x 32 m (rows >=27 zeroed)
    for (int i = tid; i < 512; i += blockDim.x) {
      const int o = i >> 4, kpl = i & 15;
      v2f wv = {0.f, 0.f};
      if (o < 27)
        wv = *reinterpret_cast<const v2f*>(ow + (size_t)o * KDIM + kc * CK + kpl * 2);
      *reinterpret_cast<v2f*>(wc + kpl * 64 + o * 2) = wv;
    }
    __syncthreads();
#pragma unroll
    for (int s8 = 0; s8 < 8; ++s8) {
      const int kpl = 2 * s8 + hi;
      const v2f a = *reinterpret_cast<const v2f*>(wc + kpl * 64 + (mtile * 16 + lo) * 2);
      const int kpg = kc * 16 + kpl;
      const v2f bb = *reinterpret_cast<const v2f*>(vl + (kpg << 7) + (nsub * 16 + lo) * 2);
      acc = __builtin_amdgcn_wmma_f32_16x16x4_f32(false, a, false, bb,
                                                  (short)0, acc, false, false);
    }
    __syncthreads();
  }

  // ---- repack + bias + sigmoid store ----
  const int p = p0 + nsub * 16 + lo;
#pragma unroll
  for (int r = 0; r < 8; ++r) {
    const int o = mtile * 16 + hi * 8 + r;
    if (o < 27) {
      float v = acc[r] + ob[o];
      int t, comp;
      if (o < 18) { t = o >> 1; comp = o & 1; }
      else        { t = o - 18; comp = 2; v = 1.f / (1.f + expf(-v)); }
      offm[(size_t)p * 27 + t * 3 + comp] = v;
    }
  }
}

// ---------------------------------------------------------------------------
// Kernel 2: deformable conv. Bilinear-sample 9 taps x 64 ch into LDS (masked),
// then WMMA GEMM: out[o][p] = sum_k W[o][k] * val[k][p], K = c*9+t.
// 256 threads = 8 waves = 4 M-tiles x 2 N-halves (each wave: 2 accumulators).
// ---------------------------------------------------------------------------
__global__ void k_deform_conv(const float* __restrict__ xh,
                              const float* __restrict__ offm,
                              const float* __restrict__ wt,    // (64,576)
                              const float* __restrict__ bias,  // (64,)
                              float* __restrict__ out) {
  extern __shared__ float smem[];
  float* vl = smem;                 // VL_FLOATS
  float* wc = smem + VL_FLOATS;     // WC2_FLOATS
  const int p0  = blockIdx.x * TILE_N;
  const int tid = threadIdx.x;

  // ---- bilinear sampling into LDS ----
  for (int item = tid; item < 9 * TILE_N; item += blockDim.x) {
    const int px = item & 63;
    const int t  = item >> 6;
    const int ti = t / 3, tj = t - 3 * (t / 3);
    const int p  = p0 + px;
    const int b  = p / HW;
    const int hw = p - b * HW;
    const int h  = hw / WWID, w = hw - (hw / WWID) * WWID;

    const float dy = offm[(size_t)p * 27 + t * 3 + 0];
    const float dx = offm[(size_t)p * 27 + t * 3 + 1];
    const float mk = offm[(size_t)p * 27 + t * 3 + 2];

    const float py  = dy + (float)(h + ti - 1);
    const float pxf = dx + (float)(w + tj - 1);
    const float y0f = floorf(py), x0f = floorf(pxf);
    const float fy = py - y0f, fx = pxf - x0f;
    const int y0 = (int)y0f, x0 = (int)x0f;
    const int y1 = y0 + 1,   x1 = x0 + 1;
    const float vy0 = (y0 >= 0 && y0 < HH)   ? 1.f : 0.f;
    const float vy1 = (y1 >= 0 && y1 < HH)   ? 1.f : 0.f;
    const float vx0 = (x0 >= 0 && x0 < WWID) ? 1.f : 0.f;
    const float vx1 = (x1 >= 0 && x1 < WWID) ? 1.f : 0.f;
    const float w00 = (1.f - fy) * (1.f - fx) * vy0 * vx0 * mk;
    const float w01 = (1.f - fy) * fx         * vy0 * vx1 * mk;
    const float w10 = fy         * (1.f - fx) * vy1 * vx0 * mk;
    const float w11 = fy         * fx         * vy1 * vx1 * mk;
    const int yc0 = min(max(y0, 0), HH - 1),   yc1 = min(max(y1, 0), HH - 1);
    const int xc0 = min(max(x0, 0), WWID - 1), xc1 = min(max(x1, 0), WWID - 1);
    const size_t rowb = (size_t)b * HW * 64;
    const float4* p00 = reinterpret_cast<const float4*>(xh + rowb + (size_t)(yc0 * WWID + xc0) * 64);
    const float4* p01 = reinterpret_cast<const float4*>(xh + rowb + (size_t)(yc0 * WWID + xc1) * 64);
    const float4* p10 = reinterpret_cast<const float4*>(xh + rowb + (size_t)(yc1 * WWID + xc0) * 64);
    const float4* p11 = reinterpret_cast<const float4*>(xh + rowb + (size_t)(yc1 * WWID + xc1) * 64);
#pragma unroll 4
    for (int c4 = 0; c4 < 16; ++c4) {
      const float4 a00 = p00[c4], a01 = p01[c4], a10 = p10[c4], a11 = p11[c4];
      int k = (c4 * 4) * 9 + t;
      vl[((k >> 1) << 7) + (px << 1) + (k & 1)] =
          w00 * a00.x + w01 * a01.x + w10 * a10.x + w11 * a11.x; k += 9;
      vl[((k >> 1) << 7) + (px << 1) + (k & 1)] =
          w00 * a00.y + w01 * a01.y + w10 * a10.y + w11 * a11.y; k += 9;
      vl[((k >> 1) << 7) + (px << 1) + (k & 1)] =
          w00 * a00.z + w01 * a01.z + w10 * a10.z + w11 * a11.z; k += 9;
      vl[((k >> 1) << 7) + (px << 1) + (k & 1)] =
          w00 * a00.w + w01 * a01.w + w10 * a10.w + w11 * a11.w;
    }
  }
  __syncthreads();

  const int lane = tid & 31, wave = tid >> 5;
  const int lo = lane & 15, hi = lane >> 4;
  const int mtile = wave & 3;          // cout tile 0..3
  const int nhalf = wave >> 2;         // 0..1
  const int n0 = nhalf * 16, n1 = (nhalf + 2) * 16;
  v8f acc0 = {}, acc1 = {};

  for (int kc = 0; kc < NCHUNK; ++kc) {
    // stage weight chunk: 16 kp x 64 m
    for (int i = tid; i < 1024; i += blockDim.x) {
      const int o = i >> 4, kpl = i & 15;
      const v2f wv = *reinterpret_cast<const v2f*>(wt + (size_t)o * KDIM + kc * CK + kpl * 2);
      *reinterpret_cast<v2f*>(wc + kpl * 128 + o * 2) = wv;
    }
    __syncthreads();
#pragma unroll
    for (int s8 = 0; s8 < 8; ++s8) {
      const int kpl = 2 * s8 + hi;
      const v2f a = *reinterpret_cast<const v2f*>(wc + kpl * 128 + (mtile * 16 + lo) * 2);
      const int kpg = kc * 16 + kpl;
      const v2f b0 = *reinterpret_cast<const v2f*>(vl + (kpg << 7) + (n0 + lo) * 2);
      const v2f b1 = *reinterpret_cast<const v2f*>(vl + (kpg << 7) + (n1 + lo) * 2);
      acc0 = __builtin_amdgcn_wmma_f32_16x16x4_f32(false, a, false, b0,
                                                   (short)0, acc0, false, false);
      acc1 = __builtin_amdgcn_wmma_f32_16x16x4_f32(false, a, false, b1,
                                                   (short)0, acc1, false, false);
    }
    __syncthreads();
  }

  // ---- store NCHW ----
  const int pA = p0 + n0 + lo;
  const int pB = p0 + n1 + lo;
#pragma unroll
  for (int r = 0; r < 8; ++r) {
    const int o = mtile * 16 + hi * 8 + r;
    const float bv = bias[o];
    {
      const int b = pA / HW; const int hw = pA - b * HW;
      out[(size_t)(b * 64 + o) * HW + hw] = acc0[r] + bv;
    }
    {
      const int b = pB / HW; const int hw = pB - b * HW;
      out[(size_t)(b * 64 + o) * HW + hw] = acc1[r] + bv;
    }
  }
}

// ---------------------------------------------------------------------------
extern "C" void kernel_launch(void* const* d_in, const int* in_sizes, int n_in,
                              void* d_out, int out_size, void* d_ws, size_t ws_size,
                              hipStream_t stream) {
  const float* x  = (const float*)d_in[0];   // (4,64,160,160)
  const float* ow = (const float*)d_in[1];   // (27,64,3,3)
  const float* ob = (const float*)d_in[2];   // (27,)
  const float* wt = (const float*)d_in[3];   // (64,64,3,3)
  const float* bs = (const float*)d_in[4];   // (64,)
  float* out = (float*)d_out;

  float* xh   = (float*)d_ws;                       // NPIX*64 floats (26.2 MB)
  float* offm = xh + (size_t)NPIX * 64;             // NPIX*27 floats (11.1 MB)

  const dim3 blk(256);
  k_nchw_to_nhwc<<<dim3(NPIX / 64), blk, 0, stream>>>(x, xh);
  k_offset_conv<<<dim3(NPIX / TILE_N), blk,
                  (VL_FLOATS + WC1_FLOATS) * sizeof(float), stream>>>(xh, ow, ob, offm);
  k_deform_conv<<<dim3(NPIX / TILE_N), blk,
                  (VL_FLOATS + WC2_FLOATS) * sizeof(float), stream>>>(xh, offm, wt, bs, out);
}